// PartAttention_74783970558129
// MI455X (gfx1250) — compile-verified
//
#include <hip/hip_runtime.h>
#include <hip/hip_bf16.h>

typedef float v2f __attribute__((ext_vector_type(2)));
typedef float v8f __attribute__((ext_vector_type(8)));

#define B_   64
#define P_   576
#define D_   768
#define BM   96
#define BN   96
#define BK   16
#define NT   (P_ / BM)   // 6 tiles per dim
#define LDSS 18          // padded row stride (floats): 18r mod 64 distinct for r=0..15

// ---------------------------------------------------------------------------
// Kernel 1: row L2 norms.  One wave32 per row; 24 coalesced loads per lane.
// ---------------------------------------------------------------------------
__global__ __launch_bounds__(256) void cosnorm_rows(const float* __restrict__ x,
                                                    float* __restrict__ norms) {
    const int wave = threadIdx.x >> 5;
    const int lane = threadIdx.x & 31;
    const int row  = blockIdx.x * 8 + wave;          // grid = (B_*P_)/8 exactly
    const float* p = x + (size_t)row * D_;
    float s = 0.0f;
    #pragma unroll
    for (int i = 0; i < D_ / 32; ++i) {
        float v = p[lane + 32 * i];
        s += v * v;
    }
    #pragma unroll
    for (int off = 16; off > 0; off >>= 1)
        s += __shfl_xor(s, off, 32);
    if (lane == 0) norms[row] = sqrtf(s);
}

// ---------------------------------------------------------------------------
// Kernel 2: per-batch C = X*X^T via V_WMMA_F32_16X16X4_F32, upper-triangular
// tile grid only (symmetry), mirrored tile stored transposed via float4.
// 192 threads = 6 waves, wave grid 3(M) x 2(N), each wave: 32x48 outputs.
// ---------------------------------------------------------------------------
__global__ __launch_bounds__(192) void cosine_gemm(const float* __restrict__ x,
                                                   const float* __restrict__ norms,
                                                   float* __restrict__ out) {
    // Decode upper-triangular tile pair (ti <= tj) from blockIdx.x in [0, 21)
    int rem = blockIdx.x;
    int ti = 0;
    while (rem >= (NT - ti)) { rem -= (NT - ti); ++ti; }
    const int tj = ti + rem;

    const int b  = blockIdx.y;
    const int m0 = ti * BM;           // row-tile base
    const int n0 = tj * BN;           // col-tile base
    const bool offdiag = (ti != tj);

    const float* X  = x + (size_t)b * P_ * D_;
    const float* NB = norms + (size_t)b * P_;

    __shared__ float As[BM * LDSS];
    __shared__ float Bs[BN * LDSS];

    const int tid  = threadIdx.x;
    const int lane = tid & 31;
    const int wave = tid >> 5;       // 0..5
    const int wm   = wave % 3;       // M-wave: rows wm*32 .. wm*32+31
    const int wn   = wave / 3;       // N-wave: cols wn*48 .. wn*48+47

    const int rlane = lane & 15;         // fragment row/col within 16
    const int khalf = (lane >> 4) * 2;   // K offset 0 or 2 per ISA A/B layout

    v8f acc[2][3];
    #pragma unroll
    for (int tm = 0; tm < 2; ++tm)
        #pragma unroll
        for (int tn = 0; tn < 3; ++tn)
            acc[tm][tn] = (v8f){0.f,0.f,0.f,0.f,0.f,0.f,0.f,0.f};

    for (int k0 = 0; k0 < D_; k0 += BK) {
        // Stage A (rows m0..m0+95) and B (rows n0..n0+95) K-slices into LDS.
        // 96 rows * 4 float4 = 384 float4 per buffer; 192 threads -> 2 each.
        float4 ar[2], br[2];
        int row[2], c4[2];
        #pragma unroll
        for (int i = 0; i < 2; ++i) {
            int idx = tid + i * 192;
            row[i] = idx >> 2;
            c4[i]  = (idx & 3) * 4;
            ar[i] = *(const float4*)(X + (size_t)(m0 + row[i]) * D_ + k0 + c4[i]);
            br[i] = *(const float4*)(X + (size_t)(n0 + row[i]) * D_ + k0 + c4[i]);
        }
        __syncthreads();
        #pragma unroll
        for (int i = 0; i < 2; ++i) {
            float* da = &As[row[i] * LDSS + c4[i]];
            float* db = &Bs[row[i] * LDSS + c4[i]];
            *(float2*)(da)     = make_float2(ar[i].x, ar[i].y);
            *(float2*)(da + 2) = make_float2(ar[i].z, ar[i].w);
            *(float2*)(db)     = make_float2(br[i].x, br[i].y);
            *(float2*)(db + 2) = make_float2(br[i].z, br[i].w);
        }
        __syncthreads();

        // 4 K-steps of 4 over the staged BK=16 slice.
        #pragma unroll
        for (int kk = 0; kk < 4; ++kk) {
            const int col = kk * 4 + khalf;      // this lane's K pair start
            v2f af[2], bf[3];
            #pragma unroll
            for (int tm = 0; tm < 2; ++tm)
                af[tm] = *(const v2f*)&As[(wm * 32 + tm * 16 + rlane) * LDSS + col];
            #pragma unroll
            for (int tn = 0; tn < 3; ++tn)
                bf[tn] = *(const v2f*)&Bs[(wn * 48 + tn * 16 + rlane) * LDSS + col];
            #pragma unroll
            for (int tm = 0; tm < 2; ++tm)
                #pragma unroll
                for (int tn = 0; tn < 3; ++tn)
                    acc[tm][tn] = __builtin_amdgcn_wmma_f32_16x16x4_f32(
                        /*neg_a=*/false, af[tm], /*neg_b=*/false, bf[tn],
                        /*c_mod=*/(short)0, acc[tm][tn],
                        /*reuse_a=*/false, /*reuse_b=*/false);
        }
    }

    // Epilogue. D layout: lane<16: N=lane, M=mbase+e; lane>=16: M=8+e.
    // v = dot * rcp(norm_r * norm_c + eps).  Normal store: per-element b32
    // (16 lanes -> 64B row segments).  Mirrored store: 8 consecutive M values
    // per lane -> two float4 (global_store_b128) at the transposed location.
    const int nlane = lane & 15;
    const int mbase = (lane >> 4) * 8;

    #pragma unroll
    for (int tm = 0; tm < 2; ++tm) {
        const int growbase = m0 + wm * 32 + tm * 16 + mbase;
        float rn[8];
        #pragma unroll
        for (int e = 0; e < 8; ++e)
            rn[e] = NB[growbase + e];
        #pragma unroll
        for (int tn = 0; tn < 3; ++tn) {
            const int gcol = n0 + wn * 48 + tn * 16 + nlane;
            const float cn = NB[gcol];
            float v[8];
            #pragma unroll
            for (int e = 0; e < 8; ++e)
                v[e] = acc[tm][tn][e] *
                       __builtin_amdgcn_rcpf(rn[e] * cn + 1e-8f);
            // normal orientation: out[b][growbase+e][gcol]
            #pragma unroll
            for (int e = 0; e < 8; ++e)
                out[((size_t)b * P_ + growbase + e) * P_ + gcol] = v[e];
            // mirrored tile for off-diagonal blocks: out[b][gcol][growbase..+7]
            if (offdiag) {
                float* pm = out + ((size_t)b * P_ + gcol) * P_ + growbase;
                *(float4*)(pm)     = make_float4(v[0], v[1], v[2], v[3]);
                *(float4*)(pm + 4) = make_float4(v[4], v[5], v[6], v[7]);
            }
        }
    }
}

extern "C" void kernel_launch(void* const* d_in, const int* in_sizes, int n_in,
                              void* d_out, int out_size, void* d_ws, size_t ws_size,
                              hipStream_t stream) {
    const float* x   = (const float*)d_in[0];
    float* out       = (float*)d_out;
    float* norms     = (float*)d_ws;      // B_*P_ floats = 147456 B

    cosnorm_rows<<<(B_ * P_) / 8, 256, 0, stream>>>(x, norms);

    dim3 grid(NT * (NT + 1) / 2, B_);     // (21, 64): upper-triangular tiles
    cosine_gemm<<<grid, 192, 0, stream>>>(x, norms, out);
}